// Grid_23270132810301
// MI455X (gfx1250) — compile-verified
//
#include <hip/hip_runtime.h>
#include <math.h>
#include <stdint.h>

// ---------------------------------------------------------------------------
// Multiresolution hash-grid encode (Instant-NGP style), N points x 16 levels.
// Gather-bound: 64 MB of tables live in MI455X's 192 MB L2. Levels 0..14 have
// numerically unambiguous resolutions -> baked in as compile-time constants
// (immediates, no kernarg streaming). Level 15 is exactly 2048*(1+-eps) under
// numpy's double chain, so its scale/upper come from host libm at launch.
// Writeback: LDS transpose + global_store_async_from_lds_b128 (ASYNCcnt path).
// ---------------------------------------------------------------------------

#define NUM_LVL    16
#define HASH_POWER 19
#define MAX_ENTRY  (1u << HASH_POWER)   // 524288
#define BLK        256
#define LSTRIDE    36                   // floats per point-row in LDS (32 data + 4 pad -> 144B, 16B aligned)

// floor(16 * b^l), b = exp(ln(128)/15); margins >= 3.5e-3 for l<=14 (safe vs ~1e-12 libm error).
// Level 15 (index unused here) is ambiguous (2047 vs 2048) -> runtime.
__device__ constexpr int RESC[16] = {16, 22, 30, 42, 58, 80, 111, 153, 212, 294,
                                     406, 561, 776, 1072, 1482, 0 /*runtime*/};
// dense <=> res^3 < 2^19 <=> res <= 80  (levels 0..5)

#if defined(__HIP_DEVICE_COMPILE__) && defined(__gfx1250__) && \
    __has_builtin(__builtin_amdgcn_global_store_async_from_lds_b128)
#define USE_ASYNC_STORE 1
typedef int v4i __attribute__((vector_size(16)));
typedef __attribute__((address_space(1))) v4i v4i_g;   // global
typedef __attribute__((address_space(3))) v4i v4i_l;   // LDS
#else
#define USE_ASYNC_STORE 0
#endif

__global__ __launch_bounds__(BLK)
void Grid_hash_encode_kernel(const float* __restrict__ x,
                             const float* __restrict__ tables,
                             float* __restrict__ out,
                             int n, float scale15, float upper15)
{
    __shared__ float smem[BLK * LSTRIDE];

    const int t = threadIdx.x;
    const long long p = (long long)blockIdx.x * BLK + t;

    float px = 0.f, py = 0.f, pz = 0.f;
    if (p < n) {
        const float* xp = x + 3 * p;
        px = xp[0]; py = xp[1]; pz = xp[2];
    }

    // Reference OFFSETS rows (index-matched to the reference's weight list):
    // (0,0,0)(0,1,0)(0,0,1)(0,1,1)(1,0,0)(1,0,1)(1,1,0)(1,1,1)
    const int O0[8] = {0,0,0,0,1,1,1,1};
    const int O1[8] = {0,1,0,1,0,0,1,1};
    const int O2[8] = {0,0,1,1,0,1,0,1};

#pragma unroll
    for (int l = 0; l < NUM_LVL; ++l) {
        const bool last  = (l == NUM_LVL - 1);
        const bool dense = (l < 6);                 // res <= 80  ->  res^3 < 2^19
        // Compile-time IEEE folds for l<15; runtime (host-libm-exact) for l==15.
        const float scale = last ? scale15 : (float)((double)RESC[l] - 1.0);
        const float upper = last ? upper15 : (float)((double)RESC[l] - 1.0001);

        // coord = clip(x * (res-1), 0, res-1.0001), all in f32 like the reference
        float cx = fminf(fmaxf(px * scale, 0.f), upper);
        float cy = fminf(fmaxf(py * scale, 0.f), upper);
        float cz = fminf(fmaxf(pz * scale, 0.f), upper);
        float fx = floorf(cx), fy = floorf(cy), fz = floorf(cz);
        int   ix = (int)fx,   iy = (int)fy,   iz = (int)fz;
        float dx = cx - fx,   dy = cy - fy,   dz = cz - fz;
        float mx = 1.f - dx,  my = 1.f - dy,  mz = 1.f - dz;

        // Weight list exactly as the reference (bit0<->x-frac, bit1<->y, bit2<->z)
        float t00 = mx * my, t01 = dx * my, t10 = mx * dy, t11 = dx * dy;
        float w[8] = { t00*mz, t01*mz, t10*mz, t11*mz,
                       t00*dz, t01*dz, t10*dz, t11*dz };

        uint32_t idx[8];
        if (dense) {
            const int r = RESC[l], r2 = RESC[l] * RESC[l];   // immediates after unroll
            const int base = ix + iy * r + iz * r2;
#pragma unroll
            for (int c = 0; c < 8; ++c)
                idx[c] = (uint32_t)(base + O0[c] + O1[c]*r + O2[c]*r2);
        } else {
            // uint32 products == low 32 bits of int64 products; mask keeps low 19
            const uint32_t h0a = (uint32_t)ix       * 3367900313u;
            const uint32_t h0b = (uint32_t)(ix + 1) * 3367900313u;
            const uint32_t h1a = (uint32_t)iy       * 2654435761u;
            const uint32_t h1b = (uint32_t)(iy + 1) * 2654435761u;
            const uint32_t h2a = (uint32_t)iz       * 805459861u;
            const uint32_t h2b = (uint32_t)(iz + 1) * 805459861u;
#pragma unroll
            for (int c = 0; c < 8; ++c)
                idx[c] = ((O0[c] ? h0b : h0a) ^
                          (O1[c] ? h1b : h1a) ^
                          (O2[c] ? h2b : h2a)) & (MAX_ENTRY - 1u);
        }

        const float2* __restrict__ tab = (const float2*)tables + (size_t)l * MAX_ENTRY;
        float2 f[8];
#pragma unroll
        for (int c = 0; c < 8; ++c) f[c] = tab[idx[c]];   // 8x global_load_b64 gathers (L2-resident)

        float a0 = 0.f, a1 = 0.f;
#pragma unroll
        for (int c = 0; c < 8; ++c) {
            a0 = fmaf(w[c], f[c].x, a0);
            a1 = fmaf(w[c], f[c].y, a1);
        }

        float2 acc; acc.x = a0; acc.y = a1;
        *(float2*)&smem[t * LSTRIDE + 2 * l] = acc;       // conflict-free ds_store_b64
    }

    __syncthreads();

    // Coalesced writeback: block region = 256 points * 32 floats = 2048 float4.
    // Thread t handles float4 chunk g = k*256 + t (consecutive lanes ->
    // consecutive 16B of global memory), sourced straight from LDS via the
    // gfx1250 async store (no VGPR staging, tracked by ASYNCcnt).
    const long long blk_base = (long long)blockIdx.x * BLK;
#pragma unroll
    for (int k = 0; k < 8; ++k) {
        const int pl = k * 32 + (t >> 3);   // local point 0..255
        const int j  = t & 7;               // float4 chunk within the 32-float row
        const long long gp = blk_base + pl;
        if (gp < n) {
            float*       gdst = out + gp * 32 + (long long)j * 4;
            const float* lsrc = &smem[pl * LSTRIDE + j * 4];   // 16B-aligned (stride 144B)
#if USE_ASYNC_STORE
            __builtin_amdgcn_global_store_async_from_lds_b128(
                (v4i_g*)(void*)gdst, (v4i_l*)(void*)lsrc, 0, 0);
#else
            *(float4*)gdst = *(const float4*)lsrc;
#endif
        }
    }
    // S_ENDPGM performs an implicit wait-idle (covers ASYNCcnt/STOREcnt).
}

extern "C" void kernel_launch(void* const* d_in, const int* in_sizes, int n_in,
                              void* d_out, int out_size, void* d_ws, size_t ws_size,
                              hipStream_t stream)
{
    const float* x      = (const float*)d_in[0];   // [N,3] f32
    const float* tables = (const float*)d_in[1];   // [16, 2^19, 2] f32
    float*       out    = (float*)d_out;           // [N, 32] f32
    const int n = in_sizes[0] / 3;

    // Level 15 resolution is mathematically exactly 2048; numpy's double chain
    // may round either side. Mirror it bit-for-bit with the same libm calls.
    const double b   = exp((log(2048.0) - log(16.0)) / 15.0);
    const double r15 = floor(16.0 * pow(b, 15.0));
    const float scale15 = (float)(r15 - 1.0);
    const float upper15 = (float)(r15 - 1.0001);

    const int blocks = (n + BLK - 1) / BLK;
    hipLaunchKernelGGL(Grid_hash_encode_kernel, dim3(blocks), dim3(BLK), 0, stream,
                       x, tables, out, n, scale15, upper15);
}